// LSTMMultiLayer_65180423684731
// MI455X (gfx1250) — compile-verified
//
#include <hip/hip_runtime.h>

// Problem constants (fixed by the reference).
#define LSTM_B 256
#define LSTM_T 2048
#define LSTM_H 32
#define LSTM_G 128   // 4*H gate columns: [i | f | g | o]
#define CHUNK  64    // producer->consumer publish granularity (time steps)
#define NCHUNK (LSTM_T / CHUNK)
#define PF_AHEAD 8   // x-stream prefetch distance (time steps)

typedef __attribute__((ext_vector_type(2))) float v2f;
typedef __attribute__((ext_vector_type(8))) float v8f;

// D(16x16,f32) = A(16x4,f32) * B(4x16,f32) + C  — CDNA5 V_WMMA_F32_16X16X4_F32
__device__ __forceinline__ v8f wmma_f32_16x16x4(v2f a, v2f b, v8f c) {
  return __builtin_amdgcn_wmma_f32_16x16x4_f32(
      /*neg_a=*/false, a, /*neg_b=*/false, b,
      /*c_mod=*/(short)0, c, /*reuse_a=*/false, /*reuse_b=*/false);
}

// Fast gate math: v_exp_f32 + v_rcp_f32, correct saturation at +/-inf.
__device__ __forceinline__ float sig_(float x) {
  return __builtin_amdgcn_rcpf(1.0f + __expf(-x));
}
__device__ __forceinline__ float tanh_(float x) {
  // tanh(x) = 1 - 2/(exp(2x)+1)
  return __builtin_fmaf(-2.0f, __builtin_amdgcn_rcpf(1.0f + __expf(2.0f * x)), 1.0f);
}

// One LSTM layer over the full sequence for one 16-row batch tile.
// 8 wave32's: wave w owns gate columns [16w,16w+16) (WMMA N-tile).
// Chunked producer/consumer pipelining across layers via AGENT-scope flags.
template <int D, bool WAIT_PRODUCER, bool PUBLISH>
__device__ __forceinline__ void lstm_layer_run(
    const float* __restrict__ X,     // [B][T][D]
    const float* __restrict__ Wih,   // [128][D]
    const float* __restrict__ Whh,   // [128][32]
    const float* __restrict__ bias,  // [128]
    float* __restrict__ Hout,        // [B][T][32]
    const int* prod_flag,            // producer progress (chunks) for this tile
    int* my_flag,                    // our progress flag (if PUBLISH)
    int tile, float* g_lds, float* h_lds)
{
  const int tid  = threadIdx.x;
  const int lane = tid & 31;
  const int wv   = tid >> 5;               // wave id 0..7 -> gate N-tile
  const int l16  = lane & 15;
  const int hoff = (lane < 16) ? 0 : 1;    // A/B K split, C row split (+8)
  const int b0   = tile * 16;
  const int ncol = wv * 16 + l16;          // gate column 0..127

  // Register-resident B fragments of Wih^T and Whh^T: (K=k,N=ncol)=W[ncol][k].
  v2f bwi[D / 4];
#pragma unroll
  for (int kk = 0; kk < D / 4; ++kk) {
    const int kc = kk * 4 + hoff * 2;
    bwi[kk].x = Wih[ncol * D + kc];
    bwi[kk].y = Wih[ncol * D + kc + 1];
  }
  v2f bwh[8];
#pragma unroll
  for (int kk = 0; kk < 8; ++kk) {
    const int kc = kk * 4 + hoff * 2;
    bwh[kk].x = Whh[ncol * LSTM_H + kc];
    bwh[kk].y = Whh[ncol * LSTM_H + kc + 1];
  }
  const float bb = bias[ncol];

  // A fragments of h_{t-1}; zero initial state.
  v2f ha[8];
#pragma unroll
  for (int kk = 0; kk < 8; ++kk) { ha[kk].x = 0.0f; ha[kk].y = 0.0f; }

  // Per-thread cell state: thread (wv,lane) owns cells (wv,lane),(wv+8,lane).
  float c_lo = 0.0f, c_hi = 0.0f;

  const float* xrow = X + (size_t)(b0 + l16) * LSTM_T * D;  // A rows (M=l16)

  for (int ch = 0; ch < NCHUNK; ++ch) {
    const int t0    = ch * CHUNK;
    const int tlast = t0 + CHUNK - 1;

    if constexpr (WAIT_PRODUCER) {
      // Acquire: invalidates WGP-level caches so producer's h is re-fetched.
      while (__hip_atomic_load(prod_flag, __ATOMIC_ACQUIRE,
                               __HIP_MEMORY_SCOPE_AGENT) <= ch) {
        __builtin_amdgcn_s_sleep(2);
      }
    }

    // Chunk prologue: pipelined input projection cx = bias + x_{t0} @ Wih^T.
    v8f cx = {bb, bb, bb, bb, bb, bb, bb, bb};
    {
      const float* xp = xrow + (size_t)t0 * D + hoff * 2;
#pragma unroll
      for (int kk = 0; kk < D / 4; ++kk) {
        v2f a;
        a.x = xp[kk * 4];
        a.y = xp[kk * 4 + 1];
        cx = wmma_f32_16x16x4(a, bwi[kk], cx);
      }
    }

    for (int t = t0; t <= tlast; ++t) {
      // ---- critical recurrent chain: g = cx + h_{t-1} @ Whh^T (2-way) ----
      v8f c  = cx;
      v8f c2 = {0.f, 0.f, 0.f, 0.f, 0.f, 0.f, 0.f, 0.f};
#pragma unroll
      for (int kk = 0; kk < 4; ++kk) c  = wmma_f32_16x16x4(ha[kk],     bwh[kk],     c);
#pragma unroll
      for (int kk = 0; kk < 4; ++kk) c2 = wmma_f32_16x16x4(ha[kk + 4], bwh[kk + 4], c2);
#pragma unroll
      for (int r = 0; r < 8; ++r) c[r] += c2[r];

      // Publish gate tile: c[r] holds g[row = r + 8*hoff][ncol].
#pragma unroll
      for (int r = 0; r < 8; ++r) g_lds[(r + hoff * 8) * 132 + ncol] = c[r];

      // ---- off-critical-path: prefetch + input projection for next step ----
      {
        const int tpf = (t + PF_AHEAD < LSTM_T) ? (t + PF_AHEAD) : (LSTM_T - 1);
        __builtin_prefetch(xrow + (size_t)tpf * D + hoff * (D / 2), 0, 3);
        const int tn = (t < tlast) ? (t + 1) : tlast;  // stay inside chunk
        const float* xp = xrow + (size_t)tn * D + hoff * 2;
        v8f cn = {bb, bb, bb, bb, bb, bb, bb, bb};
#pragma unroll
        for (int kk = 0; kk < D / 4; ++kk) {
          v2f a;
          a.x = xp[kk * 4];
          a.y = xp[kk * 4 + 1];
          cn = wmma_f32_16x16x4(a, bwi[kk], cn);
        }
        cx = cn;  // completes in the shadow of barriers/elementwise below
      }

      __syncthreads();

      // Gate nonlinearities + state update for the two owned cells.
#pragma unroll
      for (int half = 0; half < 2; ++half) {
        const int row = wv + half * 8;
        const float* g = &g_lds[row * 132 + lane];
        const float gi = sig_(g[0]);
        const float gf = sig_(g[32]);
        const float gg = tanh_(g[64]);
        const float go = sig_(g[96]);
        float& cs = half ? c_hi : c_lo;
        cs = gf * cs + gi * gg;
        const float h = go * tanh_(cs);
        h_lds[row * 33 + lane] = h;
        Hout[((size_t)(b0 + row) * LSTM_T + t) * LSTM_H + lane] = h;
      }
      __syncthreads();

      // Rebuild A fragments of h for the next step.
#pragma unroll
      for (int kk = 0; kk < 8; ++kk) {
        const int kc = kk * 4 + hoff * 2;
        ha[kk].x = h_lds[l16 * 33 + kc];
        ha[kk].y = h_lds[l16 * 33 + kc + 1];
      }
    }

    if constexpr (PUBLISH) {
      // Release: every thread drains its stores to the coherence point, then
      // one thread bumps the per-tile progress flag.
      __threadfence();
      __syncthreads();
      if (tid == 0)
        __hip_atomic_store(my_flag, ch + 1, __ATOMIC_RELEASE,
                           __HIP_MEMORY_SCOPE_AGENT);
    }
  }
}

// Wavefront pipeline over (layer, time): 48 WGs = 3 layers x 16 batch tiles.
// Layer l+1 tile i consumes exactly what layer l tile i produces.
__global__ __launch_bounds__(256) void lstm_pipeline_kernel(
    const float* __restrict__ x,
    const float* __restrict__ Wih0, const float* __restrict__ Whh0,
    const float* __restrict__ bias0,
    const float* __restrict__ Wih1, const float* __restrict__ Whh1,
    const float* __restrict__ bias1,
    const float* __restrict__ Wih2, const float* __restrict__ Whh2,
    const float* __restrict__ bias2,
    float* __restrict__ h0, float* __restrict__ h1, float* __restrict__ h2,
    int* __restrict__ flags)  // [0..15]: layer0 progress, [16..31]: layer1
{
  __shared__ float g_lds[16 * 132];  // 16 rows x 128 gates, padded stride 132
  __shared__ float h_lds[16 * 33];   // 16 rows x 32 h,      padded stride 33

  const int layer = blockIdx.x >> 4;  // 0..2 (layer-0 WGs dispatched first)
  const int tile  = blockIdx.x & 15;

  if (layer == 0) {
    lstm_layer_run<64, false, true>(x, Wih0, Whh0, bias0, h0,
                                    nullptr, &flags[tile], tile, g_lds, h_lds);
  } else if (layer == 1) {
    lstm_layer_run<32, true, true>(h0, Wih1, Whh1, bias1, h1,
                                   &flags[tile], &flags[16 + tile], tile,
                                   g_lds, h_lds);
  } else {
    lstm_layer_run<32, true, false>(h1, Wih2, Whh2, bias2, h2,
                                    &flags[16 + tile], nullptr, tile,
                                    g_lds, h_lds);
  }
}

__global__ void init_flags_kernel(int* __restrict__ flags) {
  if (threadIdx.x < 32) flags[threadIdx.x] = 0;
}

// out[b] = h_last[b,:] . W_out[0,:] + b_out
__global__ __launch_bounds__(256) void out_proj_kernel(
    const float* __restrict__ Hseq, const float* __restrict__ Wout,
    const float* __restrict__ bout, float* __restrict__ out)
{
  const int b = threadIdx.x;
  const float* hp = Hseq + ((size_t)b * LSTM_T + (LSTM_T - 1)) * LSTM_H;
  float acc = bout[0];
#pragma unroll
  for (int j = 0; j < LSTM_H; ++j) acc += hp[j] * Wout[j];
  out[b] = acc;
}

extern "C" void kernel_launch(void* const* d_in, const int* in_sizes, int n_in,
                              void* d_out, int out_size, void* d_ws, size_t ws_size,
                              hipStream_t stream) {
  const float* x    = (const float*)d_in[0];
  const float* Wih0 = (const float*)d_in[1];
  const float* Whh0 = (const float*)d_in[2];
  const float* b0   = (const float*)d_in[3];
  const float* Wih1 = (const float*)d_in[4];
  const float* Whh1 = (const float*)d_in[5];
  const float* b1   = (const float*)d_in[6];
  const float* Wih2 = (const float*)d_in[7];
  const float* Whh2 = (const float*)d_in[8];
  const float* b2   = (const float*)d_in[9];
  const float* Wout = (const float*)d_in[10];
  const float* bout = (const float*)d_in[11];
  float* out = (float*)d_out;

  // Workspace: three hidden-sequence buffers (64 MiB each) + progress flags.
  const size_t hbytes = (size_t)LSTM_B * LSTM_T * LSTM_H * sizeof(float);
  float* h0 = (float*)d_ws;
  float* h1 = (float*)((char*)d_ws + hbytes);
  float* h2 = (float*)((char*)d_ws + 2 * hbytes);
  int* flags = (int*)((char*)d_ws + 3 * hbytes);

  init_flags_kernel<<<1, 32, 0, stream>>>(flags);
  lstm_pipeline_kernel<<<48, 256, 0, stream>>>(
      x, Wih0, Whh0, b0, Wih1, Whh1, b1, Wih2, Whh2, b2, h0, h1, h2, flags);
  out_proj_kernel<<<1, 256, 0, stream>>>(h2, Wout, bout, out);
}